// CombinedRotaryEmbedding_49890340110530
// MI455X (gfx1250) — compile-verified
//
#include <hip/hip_runtime.h>
#include <math.h>

typedef __attribute__((ext_vector_type(2))) float v2f;
typedef __attribute__((ext_vector_type(8))) float v8f;

#define S_LEN   4096
#define HD      64
#define NHEAD   16
#define NROT    32
#define NB      8
#define ROWS    128     // head-rows per block
#define THREADS 256     // 8 waves (wave32)
#define XP      68      // padded row stride (floats) for x / result tile in LDS
#define BP      68      // padded column stride (floats) for M^T tile in LDS

// ---------------------------------------------------------------------------
// Setup kernel 1: fold the 32 sequential Givens rotations and r_matrix into a
// single 64x64 matrix  M = G0*G1*...*G31*R,  built as Acc=R; for k=31..0:
// Acc = G_k * Acc  (left-multiply mixes ROWS i,j).  Stored transposed:
// Mt[c*64 + k] = M[k][c]  so the main kernel's LDS staging is conflict-free.
// One block of 64 threads; thread t owns column t (no races, no barriers).
// ---------------------------------------------------------------------------
__global__ void build_combined_matrix(const float* __restrict__ thetas,
                                      const float* __restrict__ r_pairs,
                                      const float* __restrict__ theta_scale,
                                      const float* __restrict__ r_matrix,
                                      float* __restrict__ Mt) {
    __shared__ float Ms[HD * HD];
    const int t = threadIdx.x;                    // 0..63, owns column t
    for (int r = 0; r < HD; ++r) Ms[r * HD + t] = r_matrix[r * HD + t];
    const float ts = theta_scale[0];
    for (int k = NROT - 1; k >= 0; --k) {
        const int i = (int)r_pairs[2 * k];
        const int j = (int)r_pairs[2 * k + 1];
        const float ang = thetas[k] * ts;
        const float c = cosf(ang), s = sinf(ang);
        const float mi = Ms[i * HD + t];
        const float mj = Ms[j * HD + t];
        if (i == j) {                              // reference collapses to xi*c
            Ms[i * HD + t] = c * mi;
        } else {
            Ms[i * HD + t] = c * mi - s * mj;
            Ms[j * HD + t] = s * mi + c * mj;
        }
    }
    for (int r = 0; r < HD; ++r) Mt[t * HD + r] = Ms[r * HD + t];
}

// ---------------------------------------------------------------------------
// Setup kernel 2: RoPE sin/cos LUT.  tab[(s*32+p)*2] = {cos, sin} of
// s*inv_freq[p].  Only 4096*32 distinct pairs; libm sinf/cosf for accurate
// range reduction at args up to 4095 (v_sin_f32's f32 1/2pi reduction would
// cost ~4e-4 there).
// ---------------------------------------------------------------------------
__global__ void build_sincos(const float* __restrict__ inv_freq,
                             float* __restrict__ tab) {
    const int id = blockIdx.x * blockDim.x + threadIdx.x;   // 0..131071
    const int s = id >> 5;
    const int p = id & 31;
    const float arg = (float)s * inv_freq[p];
    tab[2 * id]     = cosf(arg);
    tab[2 * id + 1] = sinf(arg);
}

// ---------------------------------------------------------------------------
// Main fused kernel: out = RoPE( x @ M ).  One read of x, one write of out.
// Per block: stage 128x64 x-tile + 64x64 M^T into LDS; each of the 8 waves
// computes a 16x64 tile with V_WMMA_F32_16X16X4_F32 (4 N-tiles x 16 K-steps),
// writes D back to LDS, then a coalesced pass applies RoPE from the LUT.
// ---------------------------------------------------------------------------
__global__ __launch_bounds__(THREADS)
void rotary_main(const float* __restrict__ x,
                 const float* __restrict__ Mt,
                 const float* __restrict__ tab,
                 float* __restrict__ out) {
    __shared__ float lX[ROWS * XP];   // x tile, then matmul result tile
    __shared__ float lMt[HD * BP];    // M^T : lMt[c*BP + k] = M[k][c]

    const int t = threadIdx.x;
    const size_t blockBase = (size_t)blockIdx.x * (ROWS * HD);

    // ---- stage M^T (coalesced b128 reads, linear LDS b128 stores) ----
    {
        const float4* m4 = (const float4*)Mt;
        #pragma unroll
        for (int it = 0; it < (HD * HD / 4) / THREADS; ++it) {   // 4 iters
            const int f4 = it * THREADS + t;                     // 0..1023
            const int c  = f4 >> 4;                              // 16 float4 per col
            const int k4 = (f4 & 15) << 2;
            *(float4*)&lX[0]; // no-op alias guard
            *(float4*)&lMt[c * BP + k4] = m4[f4];
        }
    }
    // ---- stage x tile (coalesced b128 reads -> padded LDS rows) ----
    {
        const float4* xv = (const float4*)(x + blockBase);
        #pragma unroll
        for (int it = 0; it < (ROWS * HD / 4) / THREADS; ++it) { // 8 iters
            const int f4  = it * THREADS + t;                    // 0..2047
            const int row = f4 >> 4;                             // 16 float4 per row
            const int c4  = (f4 & 15) << 2;
            *(float4*)&lX[row * XP + c4] = xv[f4];
        }
    }
    __syncthreads();

    // ---- per-wave 16x64 tile:  D = A(16x64) x M(64x64)  via 16x16x4 f32 WMMA
    {
        const int wave    = t >> 5;
        const int lane    = t & 31;
        const int m       = lane & 15;      // M index / N index / column index
        const int h       = lane >> 4;      // half-wave selects K pair {0,1}/{2,3}
        const int rowBase = wave * 16;

        v8f acc[4];
        #pragma unroll
        for (int n = 0; n < 4; ++n) acc[n] = (v8f){};

        #pragma unroll
        for (int kk = 0; kk < 16; ++kk) {
            const int k0 = kk * 4 + h * 2;
            // A frag (ISA 16x4 32-bit layout): lane m, v0=K(k0), v1=K(k0+1)
            const v2f a = *(const v2f*)&lX[(rowBase + m) * XP + k0];
            #pragma unroll
            for (int n = 0; n < 4; ++n) {
                // B frag: col = n*16 + m, rows k0/k0+1 (contiguous in M^T)
                const v2f b = *(const v2f*)&lMt[(n * 16 + m) * BP + k0];
                acc[n] = __builtin_amdgcn_wmma_f32_16x16x4_f32(
                    /*neg_a=*/false, a, /*neg_b=*/false, b,
                    /*c_mod=*/(short)0, acc[n],
                    /*reuse_a=*/false, /*reuse_b=*/false);
            }
        }

        // D layout: VGPR r -> row (r + 8*h), col (n*16 + m).  Write back to lX
        // (wave only touches its own 16 rows -> no barrier needed before this).
        #pragma unroll
        for (int n = 0; n < 4; ++n) {
            #pragma unroll
            for (int r = 0; r < 8; ++r) {
                lX[(rowBase + r + 8 * h) * XP + n * 16 + m] = acc[n][r];
            }
        }
    }
    __syncthreads();

    // ---- RoPE from LUT + coalesced store ----
    {
        float* outp = out + blockBase;
        #pragma unroll
        for (int it = 0; it < (ROWS * HD) / THREADS; ++it) {     // 32 iters
            const int o     = it * THREADS + t;
            const int row   = o >> 6;
            const int c     = o & 63;
            const int p     = c & 31;
            const int upper = c >> 5;
            const int rIdx  = blockIdx.x * ROWS + row;
            const int s     = (rIdx >> 4) & (S_LEN - 1);         // (rIdx/H) % S
            const v2f cs = *(const v2f*)&tab[(s * 32 + p) * 2];  // {cos, sin}
            const v2f y  = *(const v2f*)&lX[row * XP + 2 * p];   // {x1, x2}
            outp[o] = upper ? (y.x * cs.y + y.y * cs.x)          // x1*sin + x2*cos
                            : (y.x * cs.x - y.y * cs.y);         // x1*cos - x2*sin
        }
    }
}

// ---------------------------------------------------------------------------
extern "C" void kernel_launch(void* const* d_in, const int* in_sizes, int n_in,
                              void* d_out, int out_size, void* d_ws, size_t ws_size,
                              hipStream_t stream) {
    (void)in_sizes; (void)n_in; (void)out_size; (void)ws_size;
    const float* x        = (const float*)d_in[0];
    const float* thetas   = (const float*)d_in[1];
    const float* r_pairs  = (const float*)d_in[2];
    const float* tscale   = (const float*)d_in[3];
    /* d_in[4] = n_rots_scale: unused by the reference math */
    const float* r_matrix = (const float*)d_in[5];
    const float* inv_freq = (const float*)d_in[6];
    float* out = (float*)d_out;

    float* Mt  = (float*)d_ws;            // 64*64 f32   = 16 KB
    float* tab = (float*)d_ws + HD * HD;  // 4096*32*2 f32 = 1 MB

    build_combined_matrix<<<1, 64, 0, stream>>>(thetas, r_pairs, tscale, r_matrix, Mt);
    build_sincos<<<(S_LEN * 32) / 256, 256, 0, stream>>>(inv_freq, tab);

    const int nblocks = (NB * S_LEN * NHEAD) / ROWS;  // 524288 / 128 = 4096
    rotary_main<<<nblocks, THREADS, 0, stream>>>(x, Mt, tab, out);
}